// BlockChunkedRouting_10788957847688
// MI455X (gfx1250) — compile-verified
//
#include <hip/hip_runtime.h>
#include <hip/hip_bf16.h>
#include <stdint.h>

// ---- problem constants (from reference) ----
#define BATCH      8192
#define NUM_CHUNKS 8
#define IN_FEAT    4096
#define OUT_FEAT   4096
#define CIN        512
#define COUT       512

typedef __attribute__((ext_vector_type(2))) float v2f;
typedef __attribute__((ext_vector_type(4))) float v4f;
typedef __attribute__((ext_vector_type(8))) float v8f;

// ---------------------------------------------------------------------------
// Stage 1: per-(chunk, row-slab) partial sums of |x|  (deterministic tree)
// grid = (8 chunks, 64 slabs), block = 256
// ---------------------------------------------------------------------------
__global__ __launch_bounds__(256) void act_partial_kernel(
    const float* __restrict__ x, float* __restrict__ partial) {
  const int c = blockIdx.x;        // chunk 0..7
  const int p = blockIdx.y;        // slab 0..63  (128 rows each)
  const int t = threadIdx.x;       // 0..255
  const int row0 = p * 128;

  const float* base = x + (size_t)row0 * IN_FEAT + c * CIN;
  float s = 0.0f;
  for (int r = 0; r < 128; ++r) {
    const float* rp = base + (size_t)r * IN_FEAT;
    s += fabsf(rp[t]) + fabsf(rp[t + 256]);
  }

  __shared__ float red[256];
  red[t] = s;
  __syncthreads();
  for (int off = 128; off > 0; off >>= 1) {
    if (t < off) red[t] += red[t + off];
    __syncthreads();
  }
  if (t == 0) partial[c * 64 + p] = red[0];
}

// ---------------------------------------------------------------------------
// Stage 2: finalize activities (mean), write to output tail, pick top-2.
// 1 block, 64 threads. Divisor 8192*512 = 2^22 -> exact multiply.
// ---------------------------------------------------------------------------
__global__ void act_final_kernel(const float* __restrict__ partial,
                                 float* __restrict__ acts_ws,
                                 int* __restrict__ idx_ws,
                                 float* __restrict__ act_out) {
  __shared__ float acts[NUM_CHUNKS];
  const int t = threadIdx.x;
  if (t < NUM_CHUNKS) {
    float s = 0.0f;
    for (int i = 0; i < 64; ++i) s += partial[t * 64 + i];
    const float a = s * 0x1p-22f;  // / (8192*512)
    acts[t] = a;
    acts_ws[t] = a;
    act_out[t] = a;
  }
  __syncthreads();
  if (t == 0) {
    // top-2, stable (lowest index wins on ties), matches lax.top_k ordering
    int i0 = 0;
    float v0 = acts[0];
    for (int i = 1; i < NUM_CHUNKS; ++i)
      if (acts[i] > v0) { v0 = acts[i]; i0 = i; }
    int i1 = -1;
    float v1 = -__builtin_inff();
    for (int i = 0; i < NUM_CHUNKS; ++i) {
      if (i == i0) continue;
      if (acts[i] > v1) { v1 = acts[i]; i1 = i; }
    }
    idx_ws[0] = i0;
    idx_ws[1] = i1;
  }
}

// ---------------------------------------------------------------------------
// Stage 3: GEMM + scatter.
// grid = (128 M-tiles, 4 N-tiles, 8 chunks), block = 256 (8 waves, wave32).
// Block tile: 64(M) x 128(N), K staged 32 at a time, DOUBLE-BUFFERED LDS,
// filled with global_load_async_to_lds_b128 (ASYNCcnt) while computing.
// Wave w: rows (w&3)*16, cols (w>>2)*64 -> four independent 16x16 WMMA chains.
// Unselected chunks: zero-fill (non-temporal).
// ---------------------------------------------------------------------------
#define LDST 36   // padded LDS row stride in dwords: multiple of 4 (b128-aligned),
                  // 36*n mod 64 -> 16 distinct banks for 16 consecutive n.
#define TM   64
#define TN   128
#define TK   32

__device__ __forceinline__ void async_ld16(uint32_t lds_byte_off,
                                           const float* gaddr) {
  // GLOBAL_LOAD_ASYNC_TO_LDS_B128: copies 16 bytes from global to LDS at byte offset vdst, counted by ASYNCcnt
  asm volatile("global_load_async_to_lds_b128 %0, %1, off"
               :: "v"(lds_byte_off), "v"(gaddr)
               : "memory");
}
__device__ __forceinline__ void wait_async0() {
  asm volatile("s_wait_asynccnt 0" ::: "memory");
}

__device__ __forceinline__ void load_stage(
    const float* __restrict__ x, const float* __restrict__ Wc,
    int row0, int n0, int c, int k0,
    uint32_t asBase, uint32_t bsBase, int t) {
  // As: 64 rows x 32 k -> 512 b128 chunks, 2 per thread
  int q = t;
#pragma unroll
  for (int j = 0; j < 2; ++j, q += 256) {
    const int r  = q >> 3;        // 0..63
    const int k4 = (q & 7) << 2;  // 0..28
    async_ld16(asBase + (uint32_t)(r * LDST + k4) * 4u,
               x + (size_t)(row0 + r) * IN_FEAT + c * CIN + k0 + k4);
  }
  // Bs: 128 rows x 32 k -> 1024 b128 chunks, 4 per thread
  q = t;
#pragma unroll
  for (int j = 0; j < 4; ++j, q += 256) {
    const int r  = q >> 3;        // 0..127
    const int k4 = (q & 7) << 2;  // 0..28
    async_ld16(bsBase + (uint32_t)(r * LDST + k4) * 4u,
               Wc + (size_t)(n0 + r) * CIN + k0 + k4);
  }
}

__global__ __launch_bounds__(256) void gemm_scatter_kernel(
    const float* __restrict__ x, const float* __restrict__ W,
    const float* __restrict__ bias, const int* __restrict__ idx,
    float* __restrict__ out) {
  const int mt = blockIdx.x;           // 0..127
  const int nt = blockIdx.y;           // 0..3
  const int c  = blockIdx.z;           // 0..7
  const int row0 = mt * TM;
  const int n0   = nt * TN;
  const int t    = threadIdx.x;

  float* outbase = out + (size_t)row0 * OUT_FEAT + c * COUT + n0;

  const bool sel = (c == idx[0]) || (c == idx[1]);   // block-uniform

  if (!sel) {
    // zero-fill 64x128 tile, non-temporal b128 stores
    const int col4 = (t & 31) * 4;   // 0..124
    const int rr   = t >> 5;         // 0..7
    const v4f z = {0.f, 0.f, 0.f, 0.f};
    for (int r = rr; r < TM; r += 8)
      __builtin_nontemporal_store(z, (v4f*)(outbase + (size_t)r * OUT_FEAT + col4));
    return;
  }

  __shared__ float As[2][TM * LDST];   // [m][k]  x tile, double-buffered
  __shared__ float Bs[2][TN * LDST];   // [n][k]  W tile, double-buffered

  const int wave = t >> 5;             // 0..7
  const int lane = t & 31;
  const int lm   = lane & 15;
  const int lh   = lane >> 4;
  const int m0w  = (wave & 3) * 16;    // wave row offset
  const int n0w  = (wave >> 2) * 64;   // wave col offset (4 x 16-wide subtiles)
  const int kl   = 2 * lh;             // K sub-offset per f32 A/B VGPR layout

  const float* Wc = W + (size_t)c * COUT * CIN;

  // LDS byte offsets of each buffer (generic LDS pointer: offset in low 32 bits)
  const uint32_t asB[2] = {(uint32_t)(uintptr_t)&As[0][0],
                           (uint32_t)(uintptr_t)&As[1][0]};
  const uint32_t bsB[2] = {(uint32_t)(uintptr_t)&Bs[0][0],
                           (uint32_t)(uintptr_t)&Bs[1][0]};

  v8f acc0 = {}, acc1 = {}, acc2 = {}, acc3 = {};

  // prologue: stage 0 -> buffer 0
  load_stage(x, Wc, row0, n0, c, 0, asB[0], bsB[0], t);
  wait_async0();
  __syncthreads();

  const int NSTAGE = CIN / TK;  // 16
  for (int s = 0; s < NSTAGE; ++s) {
    const int cur = s & 1;
    if (s + 1 < NSTAGE)
      load_stage(x, Wc, row0, n0, c, (s + 1) * TK, asB[cur ^ 1], bsB[cur ^ 1], t);

    const float* Ap = &As[cur][(m0w + lm) * LDST];
    const float* B0 = &Bs[cur][(n0w + lm) * LDST];
    const float* B1 = B0 + 16 * LDST;
    const float* B2 = B0 + 32 * LDST;
    const float* B3 = B0 + 48 * LDST;

#pragma unroll
    for (int kk = 0; kk < TK; kk += 4) {
      v2f a, b;
      a[0] = Ap[kk + kl];  a[1] = Ap[kk + kl + 1];
      b[0] = B0[kk + kl];  b[1] = B0[kk + kl + 1];
      acc0 = __builtin_amdgcn_wmma_f32_16x16x4_f32(
          false, a, false, b, (short)0, acc0, false, false);
      b[0] = B1[kk + kl];  b[1] = B1[kk + kl + 1];
      acc1 = __builtin_amdgcn_wmma_f32_16x16x4_f32(
          false, a, false, b, (short)0, acc1, false, false);
      b[0] = B2[kk + kl];  b[1] = B2[kk + kl + 1];
      acc2 = __builtin_amdgcn_wmma_f32_16x16x4_f32(
          false, a, false, b, (short)0, acc2, false, false);
      b[0] = B3[kk + kl];  b[1] = B3[kk + kl + 1];
      acc3 = __builtin_amdgcn_wmma_f32_16x16x4_f32(
          false, a, false, b, (short)0, acc3, false, false);
    }

    wait_async0();     // next buffer landed
    __syncthreads();   // all waves done reading cur buffer
  }

  // epilogue: bias + non-temporal store.
  // C/D layout: VGPR v -> m = v + 8*lh, n = lane&15
  const float bv0 = bias[c * COUT + n0 + n0w + lm];
  const float bv1 = bias[c * COUT + n0 + n0w + 16 + lm];
  const float bv2 = bias[c * COUT + n0 + n0w + 32 + lm];
  const float bv3 = bias[c * COUT + n0 + n0w + 48 + lm];
  float* op = outbase + (size_t)m0w * OUT_FEAT + n0w + lm;
#pragma unroll
  for (int v = 0; v < 8; ++v) {
    const size_t m = (size_t)(v + 8 * lh) * OUT_FEAT;
    __builtin_nontemporal_store(acc0[v] + bv0, op + m);
    __builtin_nontemporal_store(acc1[v] + bv1, op + m + 16);
    __builtin_nontemporal_store(acc2[v] + bv2, op + m + 32);
    __builtin_nontemporal_store(acc3[v] + bv3, op + m + 48);
  }
}

// ---------------------------------------------------------------------------
extern "C" void kernel_launch(void* const* d_in, const int* in_sizes, int n_in,
                              void* d_out, int out_size, void* d_ws, size_t ws_size,
                              hipStream_t stream) {
  const float* x = (const float*)d_in[0];   // (8192, 4096) f32
  const float* W = (const float*)d_in[1];   // (8, 512, 512) f32
  const float* b = (const float*)d_in[2];   // (8, 512) f32
  float* out = (float*)d_out;               // 8192*4096 + 8 floats

  // workspace (floats): [0..7] activities, [8..9] idx (ints), [16..528) partials
  float* ws_f     = (float*)d_ws;
  int*   ws_i     = (int*)d_ws;
  float* partials = ws_f + 16;
  int*   idxp     = ws_i + 8;
  float* act_out  = out + (size_t)BATCH * OUT_FEAT;

  act_partial_kernel<<<dim3(NUM_CHUNKS, 64), 256, 0, stream>>>(x, partials);
  act_final_kernel<<<1, 64, 0, stream>>>(partials, ws_f, idxp, act_out);
  gemm_scatter_kernel<<<dim3(BATCH / TM, COUT / TN, NUM_CHUNKS), 256, 0, stream>>>(
      x, W, b, idxp, out);
}